// AttnDecoderLSTM_29463475651098
// MI455X (gfx1250) — compile-verified
//
#include <hip/hip_runtime.h>
#include <hip/hip_bf16.h>
#include <math.h>
#include <stdint.h>

#define Bsz   1024
#define A_NUM 16
#define V_NUM 36
#define SEQ   80
#define EMB   256
#define HID   512
#define FEAT  2176
#define DOT   256

#define KC     32          // K-chunk staged in LDS
#define LDA_S  36          // A panel row stride (floats): 144B, 16B-aligned, bank-spread
#define LDB_S  72          // B panel [KC][64] row stride: 288B, 16B-aligned
#define LDBT_S 36          // B^T panel [64][KC] row stride

typedef __attribute__((ext_vector_type(2))) float v2f;
typedef __attribute__((ext_vector_type(8))) float v8f;

// Async global->LDS 16-byte copy (GLOBAL_LOAD_ASYNC_TO_LDS_B128, ASYNCcnt).
// VDST carries the LDS byte offset = low 32 bits of the flat shared address.
__device__ __forceinline__ void async_cp16(const float* lds_dst, const float* gsrc) {
  uint32_t l = (uint32_t)(uintptr_t)lds_dst;
  uint64_t g = (uint64_t)(uintptr_t)gsrc;
  asm volatile("global_load_async_to_lds_b128 %0, %1, off"
               :: "v"(l), "v"(g) : "memory");
}
__device__ __forceinline__ void wait_async_all() {
  asm volatile("s_wait_asynccnt 0x0" ::: "memory");
}

// ---------------------------------------------------------------------------
// Block-tiled fp32 WMMA GEMM (V_WMMA_F32_16X16X4_F32).
// Block = 256 threads = 8 waves computing a 32x64 macro-tile (2x4 wave grid).
// K consumed in KC=32 chunks: A(32xKC) and B(KCx64) staged into LDS with
// global_load_async_to_lds_b128, synced with s_wait_asynccnt + barrier, then
// 8 WMMAs per chunk read operands from LDS.
// WMMA f32 operand layout (ISA 7.12.2): A lane%16 = M row, lanes 0-15 carry
// K=k,k+1 / lanes 16-31 K=k+2,k+3 (v2f); C/D: lane%16 = N col, VGPR i = row
// i + 8*(lane>=16).
// Up to 3 K-segments (virtual concat), 2 bias vectors, tanh epilogue.
// ---------------------------------------------------------------------------
__device__ __forceinline__ v8f gemm_seg(v8f acc,
    const float* __restrict__ A, int lda,
    const float* __restrict__ Bm, int ldb, int K, int btrans,
    int tm, int tn, int wm, int wn, int r, int h, int tid,
    float* A_s, float* B_s)
{
  for (int k0 = 0; k0 < K; k0 += KC) {
    // ---- stage A panel: 32 rows x 32 k = 1024 floats, 1 x b128 per lane ----
    {
      int row = tid >> 3, kg = (tid & 7) << 2;
      async_cp16(&A_s[row * LDA_S + kg],
                 A + (size_t)(tm + row) * lda + k0 + kg);
    }
    // ---- stage B panel: 2048 floats, 2 x b128 per lane ----
    if (btrans) {            // B stored [N,K] -> LDS as [64][KC]
#pragma unroll
      for (int j = 0; j < 2; ++j) {
        int idx = tid + j * 256;
        int n = idx >> 3, kg = (idx & 7) << 2;
        async_cp16(&B_s[n * LDBT_S + kg],
                   Bm + (size_t)(tn + n) * ldb + k0 + kg);
      }
    } else {                 // B stored [K,N] -> LDS as [KC][64]
#pragma unroll
      for (int j = 0; j < 2; ++j) {
        int idx = tid + j * 256;
        int kr = idx >> 4, ng = (idx & 15) << 2;
        async_cp16(&B_s[kr * LDB_S + ng],
                   Bm + (size_t)(k0 + kr) * ldb + tn + ng);
      }
    }
    wait_async_all();
    __syncthreads();

    // ---- 8 WMMAs from LDS ----
    const float* ar = &A_s[(wm + r) * LDA_S];
    if (btrans) {
      const float* bc = &B_s[(wn + r) * LDBT_S];
#pragma unroll
      for (int k = 0; k < KC; k += 4) {
        v2f a = *(const v2f*)(ar + k + 2 * h);
        v2f b = *(const v2f*)(bc + k + 2 * h);
        acc = __builtin_amdgcn_wmma_f32_16x16x4_f32(false, a, false, b,
                                                    (short)0, acc, false, false);
      }
    } else {
#pragma unroll
      for (int k = 0; k < KC; k += 4) {
        int kk = k + 2 * h;
        v2f a = *(const v2f*)(ar + kk);
        v2f b;
        b.x = B_s[kk * LDB_S + wn + r];
        b.y = B_s[(kk + 1) * LDB_S + wn + r];
        acc = __builtin_amdgcn_wmma_f32_16x16x4_f32(false, a, false, b,
                                                    (short)0, acc, false, false);
      }
    }
    __syncthreads();
  }
  return acc;
}

__global__ __launch_bounds__(256) void wmma_gemm3(
    int Nt64,
    const float* A0, int lda0, const float* B0, int ldb0, int K0, int bt0,
    const float* A1, int lda1, const float* B1, int ldb1, int K1, int bt1,
    const float* A2, int lda2, const float* B2, int ldb2, int K2, int bt2,
    const float* bias0, const float* bias1, float* C, int ldc, int act)
{
  __shared__ float A_s[32 * LDA_S];   // 4.6 KB
  __shared__ float B_s[32 * LDB_S];   // 9.2 KB (== 64*LDBT_S floats for B^T)

  int tid  = threadIdx.x;
  int lane = tid & 31;
  int wave = tid >> 5;
  int tm = (blockIdx.x / Nt64) << 5;         // macro-tile origin (32 rows)
  int tn = (blockIdx.x % Nt64) << 6;         // macro-tile origin (64 cols)
  int wm = (wave >> 2) << 4;                 // 0 / 16
  int wn = (wave & 3) << 4;                  // 0 / 16 / 32 / 48
  int r = lane & 15, h = lane >> 4;

  v8f acc = {};
  acc = gemm_seg(acc, A0, lda0, B0, ldb0, K0, bt0, tm, tn, wm, wn, r, h, tid, A_s, B_s);
  if (A1) acc = gemm_seg(acc, A1, lda1, B1, ldb1, K1, bt1, tm, tn, wm, wn, r, h, tid, A_s, B_s);
  if (A2) acc = gemm_seg(acc, A2, lda2, B2, ldb2, K2, bt2, tm, tn, wm, wn, r, h, tid, A_s, B_s);

  float bc = 0.f;
  if (bias0) bc += bias0[tn + wn + r];
  if (bias1) bc += bias1[tn + wn + r];
#pragma unroll
  for (int i = 0; i < 8; ++i) {
    float v = acc[i] + bc;
    if (act) v = tanhf(v);
    C[(size_t)(tm + wm + i + 8 * h) * ldc + tn + wn + r] = v;
  }
}

// ---------------------------------------------------------------------------
// Visual soft-dot attention: attn_v[b,v] = vc[b,v,:]·w[b,:], softmax over V,
// feature[b,:] = sum_v alpha_v * vc[b,v,:].  One workgroup per batch row.
// ---------------------------------------------------------------------------
__global__ __launch_bounds__(256) void visual_attn(
    const float* __restrict__ vc, const float* __restrict__ w,
    float* __restrict__ feature, float* __restrict__ alpha_v)
{
  __shared__ float w_s[FEAT];
  __shared__ float attn_s[V_NUM];
  __shared__ float alpha_s[V_NUM];
  int b = blockIdx.x, tid = threadIdx.x;
  for (int i = tid; i < FEAT; i += 256) w_s[i] = w[(size_t)b * FEAT + i];
  __syncthreads();

  int lane = tid & 31, wave = tid >> 5;
  const float* vcb = vc + (size_t)b * V_NUM * FEAT;
  for (int v = wave; v < V_NUM; v += 8) {
    const float* row = vcb + (size_t)v * FEAT;
    float s = 0.f;
    for (int f = lane; f < FEAT; f += 32) s += row[f] * w_s[f];
    for (int off = 16; off > 0; off >>= 1) s += __shfl_down(s, off, 32);
    if (lane == 0) attn_s[v] = s;
  }
  __syncthreads();

  if (tid < V_NUM) {
    float mx = -INFINITY;
    for (int v = 0; v < V_NUM; ++v) mx = fmaxf(mx, attn_s[v]);
    float den = 0.f;
    for (int v = 0; v < V_NUM; ++v) den += expf(attn_s[v] - mx);
    float a = expf(attn_s[tid] - mx) / den;
    alpha_s[tid] = a;
    alpha_v[(size_t)b * V_NUM + tid] = a;
  }
  __syncthreads();

  for (int f = tid; f < FEAT; f += 256) {
    float acc = 0.f;
#pragma unroll 4
    for (int v = 0; v < V_NUM; ++v) acc += alpha_s[v] * vcb[(size_t)v * FEAT + f];
    feature[(size_t)b * FEAT + f] = acc;
  }
}

// ---------------------------------------------------------------------------
// LSTM elementwise: gates [B,4H] (i,f,g,o) -> c1, h1
// ---------------------------------------------------------------------------
__global__ __launch_bounds__(256) void lstm_ew(
    const float* __restrict__ gates, const float* __restrict__ c0,
    float* __restrict__ h1, float* __restrict__ c1)
{
  int idx = blockIdx.x * 256 + threadIdx.x;   // B*HID threads
  int b = idx >> 9, j = idx & 511;
  const float* g = gates + ((size_t)b << 11);
  float si = 1.f / (1.f + expf(-g[j]));
  float sf = 1.f / (1.f + expf(-g[512 + j]));
  float tg = tanhf(g[1024 + j]);
  float so = 1.f / (1.f + expf(-g[1536 + j]));
  float c = sf * c0[idx] + si * tg;
  c1[idx] = c;
  h1[idx] = so * tanhf(c);
}

// ---------------------------------------------------------------------------
// Text soft-dot attention over ctx [B,SEQ,HID] with mask; outputs alpha, wctx.
// ---------------------------------------------------------------------------
__global__ __launch_bounds__(256) void text_attn(
    const float* __restrict__ ctx, const unsigned char* __restrict__ mask,
    const float* __restrict__ tgt, float* __restrict__ alpha_out,
    float* __restrict__ wctx)
{
  __shared__ float t_s[HID];
  __shared__ float attn_s[SEQ];
  __shared__ float alpha_s[SEQ];
  int b = blockIdx.x, tid = threadIdx.x;
  for (int i = tid; i < HID; i += 256) t_s[i] = tgt[(size_t)b * HID + i];
  __syncthreads();

  int lane = tid & 31, wave = tid >> 5;
  const float* cb = ctx + (size_t)b * SEQ * HID;
  for (int s = wave; s < SEQ; s += 8) {
    const float* row = cb + (size_t)s * HID;
    float d = 0.f;
    for (int k = lane; k < HID; k += 32) d += row[k] * t_s[k];
    for (int off = 16; off > 0; off >>= 1) d += __shfl_down(d, off, 32);
    if (lane == 0) attn_s[s] = mask[(size_t)b * SEQ + s] ? -INFINITY : d;
  }
  __syncthreads();

  if (tid < SEQ) {
    float mx = -INFINITY;
    for (int s = 0; s < SEQ; ++s) mx = fmaxf(mx, attn_s[s]);
    float den = 0.f;
    for (int s = 0; s < SEQ; ++s) den += expf(attn_s[s] - mx);
    float a = expf(attn_s[tid] - mx) / den;
    alpha_s[tid] = a;
    alpha_out[(size_t)b * SEQ + tid] = a;
  }
  __syncthreads();

  for (int d = tid; d < HID; d += 256) {
    float acc = 0.f;
#pragma unroll 4
    for (int s = 0; s < SEQ; ++s) acc += alpha_s[s] * cb[(size_t)s * HID + d];
    wctx[(size_t)b * HID + d] = acc;
  }
}

// ---------------------------------------------------------------------------
// Eltwise-product scoring: logit[b,a] = sum_d t[b,d]*amat[b*A+a,d]*Wso[d] + bso
// One wave per (b,a) pair.
// ---------------------------------------------------------------------------
__global__ __launch_bounds__(256) void score_k(
    const float* __restrict__ t, const float* __restrict__ amat,
    const float* __restrict__ Wso, const float* __restrict__ bso,
    float* __restrict__ logit)
{
  int lane = threadIdx.x & 31, wave = threadIdx.x >> 5;
  int p = blockIdx.x * 8 + wave;              // 16384 pairs
  int b = p >> 4;
  const float* tv = t + (size_t)b * DOT;
  const float* av = amat + (size_t)p * DOT;
  float s = 0.f;
  for (int d = lane; d < DOT; d += 32) s += tv[d] * av[d] * Wso[d];
  for (int off = 16; off > 0; off >>= 1) s += __shfl_down(s, off, 32);
  if (lane == 0) logit[p] = s + bso[0];
}

// ---------------------------------------------------------------------------
extern "C" void kernel_launch(void* const* d_in, const int* in_sizes, int n_in,
                              void* d_out, int out_size, void* d_ws, size_t ws_size,
                              hipStream_t stream) {
  const float* u_t_prev = (const float*)d_in[0];
  const float* all_u_t  = (const float*)d_in[1];
  const float* vc       = (const float*)d_in[2];
  const float* h_0      = (const float*)d_in[3];
  const float* c_0      = (const float*)d_in[4];
  const float* ctx      = (const float*)d_in[5];
  const unsigned char* ctx_mask = (const unsigned char*)d_in[6];
  const float* Wvh = (const float*)d_in[7];
  const float* bvh = (const float*)d_in[8];
  const float* Wvv = (const float*)d_in[9];
  // d_in[10] = bvv : provably irrelevant (per-row constant inside softmax)
  const float* W_ih = (const float*)d_in[11];
  const float* b_ih = (const float*)d_in[12];
  const float* W_hh = (const float*)d_in[13];
  const float* b_hh = (const float*)d_in[14];
  const float* W_in = (const float*)d_in[15];
  const float* W_out = (const float*)d_in[16];
  const float* Wsh = (const float*)d_in[17];
  const float* bsh = (const float*)d_in[18];
  const float* Wsa = (const float*)d_in[19];
  const float* bsa = (const float*)d_in[20];
  const float* Wso = (const float*)d_in[21];
  const float* bso = (const float*)d_in[22];

  float* out = (float*)d_out;
  float* h1_out     = out;                                 // [1024,512]
  float* c1_out     = out + Bsz * HID;                     // [1024,512]
  float* alpha_out  = out + 2 * Bsz * HID;                 // [1024,80]
  float* logit_out  = alpha_out + Bsz * SEQ;               // [1024,16]
  float* alphav_out = logit_out + Bsz * A_NUM;             // [1024,36]

  // workspace layout (floats), lifetime-based reuse; peak ~27.3 MB
  float* ws = (float*)d_ws;
  float* w_buf   = ws;                        // [1024,2176] (dead after visual_attn)
  float* tgt_v   = ws + 2228224;              // [1024,256]
  float* feat    = ws + 2490368;              // [1024,2176] (dead after gates gemm)
  float* gates   = ws + 4718592;              // [1024,2048] (dead after lstm_ew)
  float* tgt_t   = ws + 0;                    // [1024,512]  reuses w_buf
  float* wctx    = ws + 524288;               // [1024,512]
  float* h_tilde = ws + 1048576;              // [1024,512]
  float* tmat    = ws + 1572864;              // [1024,256]
  float* amat    = ws + 1835008;              // [16384,256] reuses feat/gates region

  const float* NP = nullptr;

  // 1) tgt_v = h_0 @ Wvh + bvh            [1024,256], K=512
  wmma_gemm3<<<128, 256, 0, stream>>>(4,
      h_0, HID, Wvh, DOT, HID, 0,  NP,0,NP,0,0,0,  NP,0,NP,0,0,0,
      bvh, NP, tgt_v, DOT, 0);

  // 2) w = tgt_v @ Wvv^T                  [1024,2176], K=256 (folded ctx_v GEMM)
  wmma_gemm3<<<1088, 256, 0, stream>>>(34,
      tgt_v, DOT, Wvv, DOT, DOT, 1,  NP,0,NP,0,0,0,  NP,0,NP,0,0,0,
      NP, NP, w_buf, FEAT, 0);

  // 3) visual attention: alpha_v + feature
  visual_attn<<<Bsz, 256, 0, stream>>>(vc, w_buf, feat, alphav_out);

  // 4) gates = [u_t_prev|feature] @ W_ih + h_0 @ W_hh + b_ih + b_hh  [1024,2048]
  wmma_gemm3<<<1024, 256, 0, stream>>>(32,
      u_t_prev, EMB,  W_ih,                 4 * HID, EMB,  0,
      feat,     FEAT, W_ih + EMB * 4 * HID, 4 * HID, FEAT, 0,
      h_0,      HID,  W_hh,                 4 * HID, HID,  0,
      b_ih, b_hh, gates, 4 * HID, 0);

  // 5) LSTM elementwise -> h_1, c_1 (outputs)
  lstm_ew<<<(Bsz * HID) / 256, 256, 0, stream>>>(gates, c_0, h1_out, c1_out);

  // 6) tgt_t = h_1 @ W_in                 [1024,512], K=512
  wmma_gemm3<<<256, 256, 0, stream>>>(8,
      h1_out, HID, W_in, HID, HID, 0,  NP,0,NP,0,0,0,  NP,0,NP,0,0,0,
      NP, NP, tgt_t, HID, 0);

  // 7) text attention: alpha (output) + wctx
  text_attn<<<Bsz, 256, 0, stream>>>(ctx, ctx_mask, tgt_t, alpha_out, wctx);

  // 8) h_tilde = tanh([wctx|h_1] @ W_out) [1024,512], K=512+512, tanh epilogue
  wmma_gemm3<<<256, 256, 0, stream>>>(8,
      wctx,   HID, W_out,             HID, HID, 0,
      h1_out, HID, W_out + HID * HID, HID, HID, 0,
      NP, 0, NP, 0, 0, 0,
      NP, NP, h_tilde, HID, 1);

  // 9) tmat = h_tilde @ Wsh + bsh         [1024,256], K=512
  wmma_gemm3<<<128, 256, 0, stream>>>(4,
      h_tilde, HID, Wsh, DOT, HID, 0,  NP,0,NP,0,0,0,  NP,0,NP,0,0,0,
      bsh, NP, tmat, DOT, 0);

  // 10) amat = all_u_t(flat) @ Wsa + bsa  [16384,256], K=256
  wmma_gemm3<<<2048, 256, 0, stream>>>(4,
      all_u_t, EMB, Wsa, DOT, EMB, 0,  NP,0,NP,0,0,0,  NP,0,NP,0,0,0,
      bsa, NP, amat, DOT, 0);

  // 11) logit = ((tmat * amat) @ Wso + bso)  [1024,16]
  score_k<<<2048, 256, 0, stream>>>(tmat, amat, Wso, bso, logit_out);
}